// LGD_12171937317460
// MI455X (gfx1250) — compile-verified
//
#include <hip/hip_runtime.h>
#include <hip/hip_bf16.h>

typedef __attribute__((ext_vector_type(2))) float v2f;
typedef __attribute__((ext_vector_type(8))) float v8f;

#define WMMA4(a, b, c) \
  __builtin_amdgcn_wmma_f32_16x16x4_f32(false, (a), false, (b), (short)0, (c), false, false)

#define BB 4
#define NP 8192
#define MTOT (BB * NP)

// ---------- order-preserving float<->uint encoding for atomic max ----------
__device__ __forceinline__ unsigned encf(float f) {
  unsigned u = __float_as_uint(f);
  return (u & 0x80000000u) ? ~u : (u | 0x80000000u);
}
__device__ __forceinline__ float decf(unsigned u) {
  u = (u & 0x80000000u) ? (u & 0x7fffffffu) : ~u;
  return __uint_as_float(u);
}
__device__ __forceinline__ float preluf(float s, float a) { return (s >= 0.f) ? s : a * s; }

// ---------------------------------------------------------------------------
__global__ void init_zero_kernel(unsigned* __restrict__ p, int n) {
  int g = blockIdx.x * 256 + threadIdx.x;
  if (g < n) p[g] = 0u;
}

// squared norms per point
__global__ void norms_kernel(const float* __restrict__ X, float* __restrict__ n2, int C, int total) {
  int g = blockIdx.x * 256 + threadIdx.x;
  if (g >= total) return;
  const float* p = X + (long)g * C;
  float s = 0.f;
  for (int c = 0; c < C; ++c) { float v = p[c]; s += v * v; }
  n2[g] = s;
}

// ---------------------------------------------------------------------------
// KNN: per block 64 rows of one batch; loop over column tiles of 64.
// Gram tile via V_WMMA_F32_16X16X4_F32 (8 waves x 2 subtiles = 16 16x16 tiles),
// dist = n2[col] - 2*G (row norm dropped: constant per row, order-invariant),
// 64 row-owner threads keep a register top-10 (insertion, stable in col order).
__global__ __launch_bounds__(256) void knn_kernel(const float* __restrict__ X,
                                                  const float* __restrict__ n2,
                                                  int* __restrict__ idx_out, int C) {
  __shared__ float Xr[64][68];
  __shared__ float Xc[64][68];
  __shared__ float Dist[64][65];
  __shared__ float n2c[64];

  const int b = blockIdx.y;
  const int row0 = blockIdx.x * 64;
  const int tid = threadIdx.x;
  const int KP = (C + 3) & ~3;
  const float* Xb = X + (long)b * NP * C;

  for (int e = tid; e < 64 * KP; e += 256) {
    int r = e / KP, c = e % KP;
    Xr[r][c] = (c < C) ? Xb[(long)(row0 + r) * C + c] : 0.f;
  }

  float bd[10]; int bi[10];
#pragma unroll
  for (int i = 0; i < 10; ++i) { bd[i] = 3.4e38f; bi[i] = 0; }

  const int lane = tid & 31, wave = tid >> 5;
  const int sr = wave >> 1;           // subtile row 0..3 (16 rows each)
  const int sc0 = (wave & 1) * 2;     // first of two col subtiles
  const int lm = lane & 15, lh = lane >> 4;

  __syncthreads();

  for (int c0 = 0; c0 < NP; c0 += 64) {
    for (int e = tid; e < 64 * KP; e += 256) {
      int r = e / KP, c = e % KP;
      Xc[r][c] = (c < C) ? Xb[(long)(c0 + r) * C + c] : 0.f;
    }
    if (tid < 64) n2c[tid] = n2[(long)b * NP + c0 + tid];
    __syncthreads();

    v8f acc0 = {}; v8f acc1 = {};
    for (int k = 0; k < KP; k += 4) {
      const int ka = k + 2 * lh;              // A/B frag: K = v + 2*(lane/16)
      v2f a, b0, b1;
      a.x = Xr[sr * 16 + lm][ka];      a.y = Xr[sr * 16 + lm][ka + 1];
      b0.x = Xc[sc0 * 16 + lm][ka];    b0.y = Xc[sc0 * 16 + lm][ka + 1];
      b1.x = Xc[(sc0 + 1) * 16 + lm][ka]; b1.y = Xc[(sc0 + 1) * 16 + lm][ka + 1];
      acc0 = WMMA4(a, b0, acc0);
      acc1 = WMMA4(a, b1, acc1);
    }
#pragma unroll
    for (int r = 0; r < 8; ++r) {             // C/D: row = r + 8*(lane/16), col = lane%16
      int rr = sr * 16 + r + 8 * lh;
      int cA = sc0 * 16 + lm, cB = (sc0 + 1) * 16 + lm;
      Dist[rr][cA] = n2c[cA] - 2.f * acc0[r];
      Dist[rr][cB] = n2c[cB] - 2.f * acc1[r];
    }
    __syncthreads();

    if (tid < 64) {
#pragma unroll 1
      for (int j = 0; j < 64; ++j) {
        float d = Dist[tid][j];
        if (d < bd[9]) {
          int p = 9;
          while (p > 0 && d < bd[p - 1]) { bd[p] = bd[p - 1]; bi[p] = bi[p - 1]; --p; }
          bd[p] = d; bi[p] = c0 + j;
        }
      }
    }
    __syncthreads();
  }

  if (tid < 64) {
    int* o = idx_out + ((long)b * NP + row0 + tid) * 10;
#pragma unroll
    for (int i = 0; i < 10; ++i) o[i] = bi[i];
  }
}

// ---------------------------------------------------------------------------
// EdgeConv1: x=[t,t] -> f=[d,d,t,t] (12ch), w1(12x64)+prelu, w2(64x64)+prelu, max over k=10.
// 4 points/block, 64 output channels/point.
__global__ __launch_bounds__(256) void edge_conv1_kernel(const float* __restrict__ T,
                                                         const int* __restrict__ idx,
                                                         const float* __restrict__ w1,
                                                         const float* __restrict__ w2,
                                                         const float* __restrict__ pa,
                                                         float* __restrict__ out) {
  __shared__ float w1s[12 * 64];
  __shared__ float w2s[64 * 64];
  __shared__ float fs[4][10][12];
  __shared__ float hs[4][10][64];
  const int tid = threadIdx.x;
  for (int e = tid; e < 12 * 64; e += 256) w1s[e] = w1[e];
  for (int e = tid; e < 64 * 64; e += 256) w2s[e] = w2[e];

  const int p = tid >> 6, o = tid & 63;
  const long g = (long)blockIdx.x * 4 + p;
  const int b = (int)(g / NP), n = (int)(g % NP);
  const float* Tb = T + (long)b * NP * 3;
  const float ti0 = Tb[n * 3 + 0], ti1 = Tb[n * 3 + 1], ti2 = Tb[n * 3 + 2];
  const int* nb = idx + g * 10;

  if (o < 10) {
    int j = nb[o];
    float d0 = Tb[j * 3 + 0] - ti0, d1 = Tb[j * 3 + 1] - ti1, d2 = Tb[j * 3 + 2] - ti2;
    float* f = fs[p][o];
    f[0] = d0; f[1] = d1; f[2] = d2; f[3] = d0; f[4] = d1; f[5] = d2;
    f[6] = ti0; f[7] = ti1; f[8] = ti2; f[9] = ti0; f[10] = ti1; f[11] = ti2;
  }
  __syncthreads();

  const float a0 = pa[0], a1 = pa[1];
  for (int kk = 0; kk < 10; ++kk) {
    float s = 0.f;
#pragma unroll
    for (int c = 0; c < 12; ++c) s += fs[p][kk][c] * w1s[c * 64 + o];
    hs[p][kk][o] = preluf(s, a0);
  }
  __syncthreads();

  float m = -3.4e38f;
  for (int kk = 0; kk < 10; ++kk) {
    float s = 0.f;
    for (int c = 0; c < 64; ++c) s += hs[p][kk][c] * w2s[c * 64 + o];
    m = fmaxf(m, preluf(s, a1));
  }
  out[g * 64 + o] = m;
}

// ---------------------------------------------------------------------------
// EdgeConv2/3: Cin=64 -> f=[nb-x, x] (128ch). TWO=1: wA+prelu, wB+prelu, max.
// TWO=0: wA+prelu, max.  wA staged in LDS (32KB); wB streamed from L2.
template <int TWO>
__global__ __launch_bounds__(256) void edge_conv_big_kernel(const float* __restrict__ Xin,
                                                            const int* __restrict__ idx,
                                                            const float* __restrict__ wA,
                                                            const float* __restrict__ wB,
                                                            const float* __restrict__ pa, int aBase,
                                                            float* __restrict__ out) {
  __shared__ float wAs[128 * 64];
  __shared__ float fs[4][10][128];
  __shared__ float hs[TWO ? 4 : 1][10][64];
  const int tid = threadIdx.x;
  for (int e = tid; e < 128 * 64; e += 256) wAs[e] = wA[e];

  const int p = tid >> 6, o = tid & 63;
  const long g = (long)blockIdx.x * 4 + p;
  const int b = (int)(g / NP), n = (int)(g % NP);
  const float* Xb = Xin + (long)b * NP * 64;
  const int* nb = idx + g * 10;

  for (int e = o; e < 10 * 128; e += 64) {
    int kk = e >> 7, c = e & 127;
    float v;
    if (c < 64) v = Xb[(long)nb[kk] * 64 + c] - Xb[(long)n * 64 + c];
    else        v = Xb[(long)n * 64 + (c - 64)];
    fs[p][kk][c] = v;
  }
  __syncthreads();

  const float aA = pa[aBase];
  float m = -3.4e38f;
  for (int kk = 0; kk < 10; ++kk) {
    float s = 0.f;
    for (int c = 0; c < 128; ++c) s += fs[p][kk][c] * wAs[c * 64 + o];
    s = preluf(s, aA);
    if (TWO) hs[p][kk][o] = s; else m = fmaxf(m, s);
  }
  if (TWO) {
    const float aB = pa[aBase + 1];
    __syncthreads();
    for (int kk = 0; kk < 10; ++kk) {
      float s = 0.f;
      for (int c = 0; c < 64; ++c) s += hs[p][kk][c] * wB[c * 64 + o];
      m = fmaxf(m, preluf(s, aB));
    }
  }
  out[g * 64 + o] = m;
}

// ---------------------------------------------------------------------------
// LDS-tiled f32 WMMA GEMM: Out[M,NC] = prelu(A[M,K] @ W[K,NC]).
// Block tile 128x64, 8 waves, each wave 16 rows x 64 cols (4 accumulators,
// A fragment reused across the 4 col subtiles). K tiled by 16 (4 k-steps).
// MODE 0: plain A.  MODE 1: A = [x1|x2|x3] (K=192).  MODE 2: + x5max bcast (K=1216).
// maxOut!=null: skip store, fused per-batch column max via encoded atomicMax.
template <int MODE>
__global__ __launch_bounds__(256) void wmma_gemm_kernel(
    const float* __restrict__ A, const float* __restrict__ x1, const float* __restrict__ x2,
    const float* __restrict__ x3, const unsigned* __restrict__ x5enc,
    const float* __restrict__ W, const float* __restrict__ pa, int aIdx,
    float* __restrict__ Out, unsigned* __restrict__ maxOut, int K, int NC) {
  __shared__ float As[128][17];
  __shared__ float Bs[16][65];
  const int tid = threadIdx.x;
  const int m0 = blockIdx.x * 128;
  const int n0 = blockIdx.y * 64;
  const int wave = tid >> 5, lane = tid & 31, lm = lane & 15, lh = lane >> 4;

  v8f acc[4] = {};

  for (int k0 = 0; k0 < K; k0 += 16) {
    for (int e = tid; e < 128 * 16; e += 256) {
      int r = e >> 4, c = e & 15;
      int row = m0 + r, k = k0 + c;
      float v;
      if (MODE == 0)      v = A[(long)row * K + k];
      else if (k < 64)    v = x1[(long)row * 64 + k];
      else if (k < 128)   v = x2[(long)row * 64 + k - 64];
      else if (k < 192)   v = x3[(long)row * 64 + k - 128];
      else                v = decf(x5enc[(row / NP) * 1024 + (k - 192)]);
      As[r][c] = v;
    }
    for (int e = tid; e < 16 * 64; e += 256) {
      int r = e >> 6, c = e & 63;
      Bs[r][c] = W[(long)(k0 + r) * NC + n0 + c];
    }
    __syncthreads();

#pragma unroll
    for (int kk = 0; kk < 16; kk += 4) {
      const int ka = kk + 2 * lh;
      v2f a;
      a.x = As[wave * 16 + lm][ka]; a.y = As[wave * 16 + lm][ka + 1];
#pragma unroll
      for (int cs = 0; cs < 4; ++cs) {
        v2f bb;
        bb.x = Bs[ka][cs * 16 + lm]; bb.y = Bs[ka + 1][cs * 16 + lm];
        acc[cs] = WMMA4(a, bb, acc[cs]);
      }
    }
    __syncthreads();
  }

  const float al = pa[aIdx];
  const int b = m0 / NP;  // 128-row tiles never straddle a batch (8192 % 128 == 0)
#pragma unroll
  for (int cs = 0; cs < 4; ++cs) {
#pragma unroll
    for (int r = 0; r < 8; ++r) {
      int row = m0 + wave * 16 + r + 8 * lh;
      int col = n0 + cs * 16 + lm;
      float s = preluf(acc[cs][r], al);
      if (maxOut) atomicMax(&maxOut[b * 1024 + col], encf(s));
      else        Out[(long)row * NC + col] = s;
    }
  }
}

// ---------------------------------------------------------------------------
// Final 128->2 layer (too narrow for WMMA): one point per thread, w10 in LDS.
__global__ __launch_bounds__(256) void final_kernel(const float* __restrict__ y9,
                                                    const float* __restrict__ w10,
                                                    const float* __restrict__ pa,
                                                    float* __restrict__ out) {
  __shared__ float w[256];
  w[threadIdx.x] = w10[threadIdx.x];
  __syncthreads();
  const int g = blockIdx.x * 256 + threadIdx.x;
  const float* r = y9 + (long)g * 128;
  const float a = pa[9];
  float s0 = 0.f, s1 = 0.f;
  for (int c = 0; c < 128; ++c) { float v = r[c]; s0 += v * w[c * 2 + 0]; s1 += v * w[c * 2 + 1]; }
  out[g * 2 + 0] = preluf(s0, a);
  out[g * 2 + 1] = preluf(s1, a);
}

// ---------------------------------------------------------------------------
extern "C" void kernel_launch(void* const* d_in, const int* in_sizes, int n_in,
                              void* d_out, int out_size, void* d_ws, size_t ws_size,
                              hipStream_t stream) {
  (void)in_sizes; (void)n_in; (void)out_size; (void)ws_size;
  const float* targets = (const float*)d_in[0];
  const float* w[10];
  for (int i = 0; i < 10; ++i) w[i] = (const float*)d_in[1 + i];
  const float* pa = (const float*)d_in[11];

  // scratch carve-out (~110 MB)
  float*    n2    = (float*)d_ws;                   // 32768
  int*      idxb  = (int*)(n2 + MTOT);              // 327680
  float*    x1    = (float*)(idxb + MTOT * 10);     // 2M
  float*    x2b   = x1 + (long)MTOT * 64;           // 2M
  float*    x3b   = x2b + (long)MTOT * 64;          // 2M
  unsigned* x5enc = (unsigned*)(x3b + (long)MTOT * 64);  // 4096
  float*    y7    = (float*)(x5enc + BB * 1024);    // 8M
  float*    y8    = y7 + (long)MTOT * 256;          // 8M
  float*    y9    = y8 + (long)MTOT * 256;          // 4M

  const dim3 blk(256);
  const dim3 knnGrid(NP / 64, BB);

  // EdgeConv1 (KNN on 3-ch targets: x=[t,t] scales distances by 2, order-invariant)
  norms_kernel<<<MTOT / 256, blk, 0, stream>>>(targets, n2, 3, MTOT);
  knn_kernel<<<knnGrid, blk, 0, stream>>>(targets, n2, idxb, 3);
  edge_conv1_kernel<<<MTOT / 4, blk, 0, stream>>>(targets, idxb, w[0], w[1], pa, x1);

  // EdgeConv2
  norms_kernel<<<MTOT / 256, blk, 0, stream>>>(x1, n2, 64, MTOT);
  knn_kernel<<<knnGrid, blk, 0, stream>>>(x1, n2, idxb, 64);
  edge_conv_big_kernel<1><<<MTOT / 4, blk, 0, stream>>>(x1, idxb, w[2], w[3], pa, 2, x2b);

  // EdgeConv3
  norms_kernel<<<MTOT / 256, blk, 0, stream>>>(x2b, n2, 64, MTOT);
  knn_kernel<<<knnGrid, blk, 0, stream>>>(x2b, n2, idxb, 64);
  edge_conv_big_kernel<0><<<MTOT / 4, blk, 0, stream>>>(x2b, idxb, w[4], nullptr, pa, 4, x3b);

  // x5 = global max over N of prelu([x1|x2|x3] @ w6): fused column-max, no x5 materialization
  init_zero_kernel<<<(BB * 1024 + 255) / 256, blk, 0, stream>>>(x5enc, BB * 1024);
  wmma_gemm_kernel<1><<<dim3(MTOT / 128, 1024 / 64), blk, 0, stream>>>(
      nullptr, x1, x2b, x3b, nullptr, w[5], pa, 5, nullptr, x5enc, 192, 1024);

  // y7 = prelu([x1|x2|x3|x5bcast] @ w7)
  wmma_gemm_kernel<2><<<dim3(MTOT / 128, 256 / 64), blk, 0, stream>>>(
      nullptr, x1, x2b, x3b, x5enc, w[6], pa, 6, y7, nullptr, 1216, 256);

  // y8 = prelu(y7 @ w8), y9 = prelu(y8 @ w9)
  wmma_gemm_kernel<0><<<dim3(MTOT / 128, 256 / 64), blk, 0, stream>>>(
      y7, nullptr, nullptr, nullptr, nullptr, w[7], pa, 7, y8, nullptr, 256, 256);
  wmma_gemm_kernel<0><<<dim3(MTOT / 128, 128 / 64), blk, 0, stream>>>(
      y8, nullptr, nullptr, nullptr, nullptr, w[8], pa, 8, y9, nullptr, 256, 128);

  // out = prelu(y9 @ w10)[..., :2]
  final_kernel<<<MTOT / 256, blk, 0, stream>>>(y9, w[9], pa, (float*)d_out);
}